// SimpleGATLayer_39144331936103
// MI455X (gfx1250) — compile-verified
//
#include <hip/hip_runtime.h>
#include <hip/hip_bf16.h>

typedef __attribute__((ext_vector_type(2))) float v2f;
typedef __attribute__((ext_vector_type(8))) float v8f;

#define F_DIM 128
#define LEAKY 0.2f

// ---------- order-preserving float<->u32 mapping for atomic max ----------
__device__ __forceinline__ unsigned f2ord(float f) {
  unsigned u = __float_as_uint(f);
  return (u & 0x80000000u) ? ~u : (u | 0x80000000u);
}
__device__ __forceinline__ float ord2f(unsigned u) {
  return (u & 0x80000000u) ? __uint_as_float(u & 0x7FFFFFFFu)
                           : __uint_as_float(~u);
}

// ---------------- Kernel 1: h = x @ W via V_WMMA_F32_16X16X4_F32 ----------
// One wave per 16x16 output tile; 8 waves/block cover all 8 N-tiles of one
// 16-row M-tile. x tile staged in LDS (padded stride 132 -> bank-conflict
// free). W is read exactly once per block (16 cols x 128 rows per wave) in
// 64B-contiguous half-wave segments, served out of WGP$/L2.
__global__ __launch_bounds__(256) void gat_gemm_wmma(
    const float* __restrict__ x, const float* __restrict__ W,
    float* __restrict__ h, int M)
{
  __shared__ float sA[16 * 132];
  const int tid = threadIdx.x;
  const int m0  = blockIdx.x * 16;

  // cooperative load of the 16x128 x-tile (512 float4)
  for (int i = tid; i < 512; i += 256) {
    int r = i >> 5, c = i & 31;
    float4 v = make_float4(0.f, 0.f, 0.f, 0.f);
    if (m0 + r < M) v = ((const float4*)(x + (size_t)(m0 + r) * F_DIM))[c];
    ((float4*)sA)[r * 33 + c] = v;
  }
  __syncthreads();

  const int lane = tid & 31;
  const int wv   = tid >> 5;          // wave id -> N tile
  const int row  = lane & 15;         // M index (A) / N index (B, C/D)
  const int kh   = (lane >> 4) * 2;   // K offset for upper half-wave
  const int n0   = wv * 16;

  v8f acc = {};
  #pragma unroll 4
  for (int k = 0; k < F_DIM; k += 4) {
    v2f a, b;
    // A 16x4 f32: lane holds M=row, VGPR v -> K = k + kh + v
    a.x = sA[row * 132 + k + kh + 0];
    a.y = sA[row * 132 + k + kh + 1];
    // B 4x16 f32: lane holds N=row, VGPR v -> K = k + kh + v
    b.x = W[(k + kh + 0) * F_DIM + n0 + row];
    b.y = W[(k + kh + 1) * F_DIM + n0 + row];
    acc = __builtin_amdgcn_wmma_f32_16x16x4_f32(false, a, false, b,
                                                (short)0, acc, false, false);
  }

  // C/D layout: N = lane&15 ; M = vgpr + 8*(lane>>4)
  const int mh = (lane >> 4) * 8;
  float* hp = h + (size_t)m0 * F_DIM + n0 + row;
  if (m0 + 16 <= M) {
    // uniform fast path (always taken for M % 16 == 0): straight-line stores
    #pragma unroll
    for (int v = 0; v < 8; ++v) hp[(size_t)(mh + v) * F_DIM] = acc[v];
  } else {
    #pragma unroll
    for (int v = 0; v < 8; ++v)
      if (m0 + mh + v < M) hp[(size_t)(mh + v) * F_DIM] = acc[v];
  }
}

// ---------------- Kernel 2: per-node attention scalars --------------------
// alpha_src[n] = h[n] . a[0:128] ; alpha_dst[n] = h[n] . a[128:256]
__global__ __launch_bounds__(256) void gat_alpha(
    const float* __restrict__ h, const float* __restrict__ a,
    float* __restrict__ as, float* __restrict__ ad, int N)
{
  const int node = blockIdx.x * 8 + (threadIdx.x >> 5);
  const int lane = threadIdx.x & 31;
  if (node >= N) return;
  float4 hv = ((const float4*)(h + (size_t)node * F_DIM))[lane];
  float4 a1 = ((const float4*)a)[lane];
  float4 a2 = ((const float4*)(a + F_DIM))[lane];
  float s = hv.x * a1.x + hv.y * a1.y + hv.z * a1.z + hv.w * a1.w;
  float d = hv.x * a2.x + hv.y * a2.y + hv.z * a2.z + hv.w * a2.w;
  #pragma unroll
  for (int m = 16; m >= 1; m >>= 1) {
    s += __shfl_xor(s, m, 32);
    d += __shfl_xor(d, m, 32);
  }
  if (lane == 0) { as[node] = s; ad[node] = d; }
}

// ---------------- Kernel 3: edge logits + segment max ---------------------
__global__ __launch_bounds__(256) void gat_edge_logit(
    const int* __restrict__ src, const int* __restrict__ dst,
    const float* __restrict__ as, const float* __restrict__ ad,
    float* __restrict__ e, unsigned* __restrict__ segmax, int E)
{
  int i = blockIdx.x * blockDim.x + threadIdx.x;
  if (i >= E) return;
  float v = as[src[i]] + ad[dst[i]];
  v = v > 0.f ? v : LEAKY * v;        // leaky_relu, alpha = 0.2
  e[i] = v;
  atomicMax(segmax + dst[i], f2ord(v));
}

// ---------------- Kernel 4: exp(e - max) + segment sum --------------------
__global__ __launch_bounds__(256) void gat_edge_exp(
    const int* __restrict__ dst, const float* __restrict__ e,
    const unsigned* __restrict__ segmax, float* __restrict__ ex,
    float* __restrict__ segsum, int E)
{
  int i = blockIdx.x * blockDim.x + threadIdx.x;
  if (i >= E) return;
  int d = dst[i];
  float v = expf(e[i] - ord2f(segmax[d]));
  ex[i] = v;
  atomicAdd(segsum + d, v);
}

// ---------------- Kernel 5: weighted scatter-add aggregation --------------
// One wave per edge; each lane owns a float4 feature slice. The edge id is
// forced into an SGPR (readfirstlane) so the per-edge scalars (src, dst,
// ex, segsum) lower to scalar loads instead of 32 redundant lane loads.
// h and out are L2-resident (25.6MB each vs 192MB L2), so the gathers and
// the f32 atomic adds are all served at L2, not HBM.
__global__ __launch_bounds__(256) void gat_aggregate(
    const int* __restrict__ src, const int* __restrict__ dst,
    const float* __restrict__ h, const float* __restrict__ ex,
    const float* __restrict__ segsum, float* __restrict__ out, int E)
{
  const int i    = __builtin_amdgcn_readfirstlane(blockIdx.x * 8 + (threadIdx.x >> 5));
  const int lane = threadIdx.x & 31;
  if (i >= E) return;
  const int s = src[i], d = dst[i];
  const float coeff = ex[i] / (segsum[d] + 1e-16f);
  float4 hv = ((const float4*)(h + (size_t)s * F_DIM))[lane];
  float* op = out + (size_t)d * F_DIM + lane * 4;
  atomicAdd(op + 0, coeff * hv.x);
  atomicAdd(op + 1, coeff * hv.y);
  atomicAdd(op + 2, coeff * hv.z);
  atomicAdd(op + 3, coeff * hv.w);
}

extern "C" void kernel_launch(void* const* d_in, const int* in_sizes, int n_in,
                              void* d_out, int out_size, void* d_ws, size_t ws_size,
                              hipStream_t stream) {
  const float* x  = (const float*)d_in[0];
  const int*   ei = (const int*)d_in[1];
  const float* W  = (const float*)d_in[2];
  const float* a  = (const float*)d_in[3];
  const int N = in_sizes[0] / F_DIM;   // 50000
  const int E = in_sizes[1] / 2;       // 800000
  const int* src = ei;
  const int* dst = ei + E;

  // workspace carve (~33 MB): h | alpha_src | alpha_dst | segmax | segsum | e | ex
  float*    h      = (float*)d_ws;
  float*    as     = h + (size_t)N * F_DIM;
  float*    ad     = as + N;
  unsigned* segmax = (unsigned*)(ad + N);
  float*    segsum = (float*)(segmax + N);
  float*    e      = segsum + N;
  float*    ex     = e + E;
  float*    out    = (float*)d_out;

  hipMemsetAsync(d_out, 0, (size_t)out_size * sizeof(float), stream);
  hipMemsetAsync(segmax, 0, (size_t)N * sizeof(unsigned), stream); // decodes below any real float
  hipMemsetAsync(segsum, 0, (size_t)N * sizeof(float), stream);

  gat_gemm_wmma <<<(N + 15) / 16,  256, 0, stream>>>(x, W, h, N);
  gat_alpha     <<<(N + 7)  / 8,   256, 0, stream>>>(h, a, as, ad, N);
  gat_edge_logit<<<(E + 255) / 256, 256, 0, stream>>>(src, dst, as, ad, e, segmax, E);
  gat_edge_exp  <<<(E + 255) / 256, 256, 0, stream>>>(dst, e, segmax, ex, segsum, E);
  gat_aggregate <<<(E + 7)  / 8,   256, 0, stream>>>(src, dst, h, ex, segsum, out, E);
}